// GATLayer_37220186587489
// MI455X (gfx1250) — compile-verified
//
#include <hip/hip_runtime.h>
#include <hip/hip_bf16.h>

#define NNODES 50000
#define NEDGES 800000
#define DCH    64
#define NHEADS 4
#define HD     (NHEADS * DCH)   // 256
#define NTOT   (NEDGES + NNODES)

typedef __attribute__((ext_vector_type(2))) float v2f;
typedef __attribute__((ext_vector_type(8))) float v8f;

// Monotone float <-> uint32 key so integer atomicMax implements float max
// (handles negatives correctly).
__device__ __forceinline__ unsigned flipf(float f) {
  unsigned u = __float_as_uint(f);
  return (u & 0x80000000u) ? ~u : (u | 0x80000000u);
}
__device__ __forceinline__ float unflipf(unsigned k) {
  unsigned u = (k & 0x80000000u) ? (k & 0x7FFFFFFFu) : ~k;
  return __uint_as_float(u);
}
#define NEG_INF_KEY 0x007FFFFFu   // flipf(-inf)

// ---------------- K0: init out / denom / segmax keys ----------------
__global__ void gat_init(float* __restrict__ out, float* __restrict__ denom,
                         unsigned* __restrict__ segkey) {
  int i = blockIdx.x * 256 + threadIdx.x;
  if (i < NNODES * DCH) out[i] = 0.f;
  if (i < NNODES * NHEADS) { denom[i] = 0.f; segkey[i] = NEG_INF_KEY; }
}

// ---------------- K1: xh = x @ W^T via V_WMMA_F32_16X16X4_F32 ----------------
// grid.x = N/16 blocks of 256 threads (8 waves). Wave w computes col tiles
// 2w and 2w+1 of the 16x256 output stripe. K=64 -> 16 WMMAs per tile.
// A(16x4) f32 layout: lanes 0-15 hold M=lane, VGPR0/1 = K=0/1; lanes 16-31 K=2/3.
// B(4x16): lane n = lane&15, VGPR0/1 = K = 2*(lane>>4) + 0/1.
// C/D(16x16): VGPR r -> row r + 8*(lane>>4), col = lane&15.
__global__ void __launch_bounds__(256)
gat_gemm_wmma(const float* __restrict__ x, const float* __restrict__ W,
              float* __restrict__ xh) {
  const int tile_m = blockIdx.x * 16;
  const int wave   = threadIdx.x >> 5;
  const int lane   = threadIdx.x & 31;
  const int mlane  = lane & 15;
  const int khalf  = (lane >> 4) << 1;           // 0 or 2
  const float* xrow = x + (size_t)(tile_m + mlane) * DCH;

  #pragma unroll
  for (int t = 0; t < 2; ++t) {
    const int c = (wave * 2 + t) * 16;           // output col base 0..240
    const float* wrow = W + (size_t)(c + mlane) * DCH;  // W[j,:], j = c + lane
    v8f acc = {};
    #pragma unroll
    for (int k = 0; k < DCH; k += 4) {
      const float2 av = *(const float2*)(xrow + k + khalf);
      const float2 bv = *(const float2*)(wrow + k + khalf);
      v2f a; a.x = av.x; a.y = av.y;
      v2f b; b.x = bv.x; b.y = bv.y;
      acc = __builtin_amdgcn_wmma_f32_16x16x4_f32(
          false, a, false, b, (short)0, acc, false, false);
    }
    const int rbase = (lane >> 4) << 3;          // 0 or 8
    float* orow = xh + (size_t)tile_m * HD + c + mlane;
    #pragma unroll
    for (int r = 0; r < 8; ++r)
      orow[(size_t)(rbase + r) * HD] = acc[r];
  }
}

// ---------------- K2: per-node attention logits a_src/a_dst [N,H] ----------------
// One wave per node: lane covers 8 consecutive channels; head = lane/8.
__global__ void __launch_bounds__(256)
gat_logits(const float* __restrict__ xh, const float* __restrict__ att_src,
           const float* __restrict__ att_dst, float* __restrict__ a_src,
           float* __restrict__ a_dst) {
  const int node = blockIdx.x * 8 + (threadIdx.x >> 5);
  const int lane = threadIdx.x & 31;
  if (node >= NNODES) return;
  const float4* xr = (const float4*)(xh + (size_t)node * HD);
  float4 p0 = xr[lane * 2];
  float4 p1 = xr[lane * 2 + 1];
  const int h = lane >> 3;
  const int dbase = (lane & 7) * 8;
  const float4* as = (const float4*)(att_src + h * DCH + dbase);
  const float4* ad = (const float4*)(att_dst + h * DCH + dbase);
  float4 s0 = as[0], s1 = as[1];
  float4 t0 = ad[0], t1 = ad[1];
  float ss = p0.x*s0.x + p0.y*s0.y + p0.z*s0.z + p0.w*s0.w
           + p1.x*s1.x + p1.y*s1.y + p1.z*s1.z + p1.w*s1.w;
  float dd = p0.x*t0.x + p0.y*t0.y + p0.z*t0.z + p0.w*t0.w
           + p1.x*t1.x + p1.y*t1.y + p1.z*t1.z + p1.w*t1.w;
  #pragma unroll
  for (int off = 4; off > 0; off >>= 1) {
    ss += __shfl_down(ss, off, 8);
    dd += __shfl_down(dd, off, 8);
  }
  if ((lane & 7) == 0) {
    a_src[node * NHEADS + h] = ss;
    a_dst[node * NHEADS + h] = dd;
  }
}

// ---------------- K3: per-edge leaky-relu score + segment max ----------------
__global__ void __launch_bounds__(256)
gat_edge_score(const int* __restrict__ ei, const float* __restrict__ a_src,
               const float* __restrict__ a_dst, float* __restrict__ alpha,
               unsigned* __restrict__ segkey) {
  const int e = blockIdx.x * 256 + threadIdx.x;
  if (e >= NTOT) return;
  int s, dn;
  if (e < NEDGES) { s = ei[e]; dn = ei[NEDGES + e]; }
  else            { s = dn = e - NEDGES; }            // self loops
  #pragma unroll
  for (int h = 0; h < NHEADS; ++h) {
    float v = a_src[s * NHEADS + h] + a_dst[dn * NHEADS + h];
    v = (v > 0.f) ? v : v * 0.2f;                     // leaky relu
    alpha[(size_t)e * NHEADS + h] = v;
    atomicMax(&segkey[dn * NHEADS + h], flipf(v));
  }
}

// ---------------- K4: exp(alpha - max) + segment sum ----------------
__global__ void __launch_bounds__(256)
gat_edge_exp(const int* __restrict__ ei, const unsigned* __restrict__ segkey,
             float* __restrict__ alpha, float* __restrict__ denom) {
  const int e = blockIdx.x * 256 + threadIdx.x;
  if (e >= NTOT) return;
  const int dn = (e < NEDGES) ? ei[NEDGES + e] : (e - NEDGES);
  #pragma unroll
  for (int h = 0; h < NHEADS; ++h) {
    float m = unflipf(segkey[dn * NHEADS + h]);
    float v = expf(alpha[(size_t)e * NHEADS + h] - m);
    alpha[(size_t)e * NHEADS + h] = v;
    atomicAdd(&denom[dn * NHEADS + h], v);
  }
}

// ---------------- K5: weighted aggregation, head-mean folded in ----------------
// One wave per edge; lane handles channels (lane) and (lane+32).
// msg[d] = 0.25 * sum_h alpha_norm[h] * xh[src, h, d]; atomic-add 64 floats/edge.
__global__ void __launch_bounds__(256)
gat_aggregate(const int* __restrict__ ei, const float* __restrict__ xh,
              const float* __restrict__ alpha, const float* __restrict__ denom,
              float* __restrict__ out) {
  const int e    = blockIdx.x * 8 + (threadIdx.x >> 5);
  const int lane = threadIdx.x & 31;
  if (e >= NTOT) return;
  int s, dn;
  if (e < NEDGES) { s = ei[e]; dn = ei[NEDGES + e]; }
  else            { s = dn = e - NEDGES; }
  float w[NHEADS];
  #pragma unroll
  for (int h = 0; h < NHEADS; ++h)
    w[h] = alpha[(size_t)e * NHEADS + h] /
           (denom[dn * NHEADS + h] + 1e-16f) * 0.25f;
  const float* xr = xh + (size_t)s * HD;
  float m0 = 0.f, m1 = 0.f;
  #pragma unroll
  for (int h = 0; h < NHEADS; ++h) {
    m0 += w[h] * xr[h * DCH + lane];
    m1 += w[h] * xr[h * DCH + lane + 32];
  }
  atomicAdd(&out[(size_t)dn * DCH + lane],      m0);
  atomicAdd(&out[(size_t)dn * DCH + lane + 32], m1);
}

// ---------------- K6: add bias ----------------
__global__ void gat_bias(float* __restrict__ out, const float* __restrict__ bias) {
  int i = blockIdx.x * 256 + threadIdx.x;
  if (i < NNODES * DCH) out[i] += bias[i & (DCH - 1)];
}

extern "C" void kernel_launch(void* const* d_in, const int* in_sizes, int n_in,
                              void* d_out, int out_size, void* d_ws, size_t ws_size,
                              hipStream_t stream) {
  const float* x       = (const float*)d_in[0];
  const int*   ei      = (const int*)  d_in[1];   // [2, E]
  const float* W       = (const float*)d_in[2];   // [256, 64]
  const float* att_src = (const float*)d_in[3];   // [4, 64]
  const float* att_dst = (const float*)d_in[4];
  const float* bias    = (const float*)d_in[5];
  float* out = (float*)d_out;

  // Workspace layout (floats): xh | a_src | a_dst | segkey | denom | alpha
  float* ws      = (float*)d_ws;
  float* xh      = ws;                                   // N*256
  float* a_src   = xh + (size_t)NNODES * HD;             // N*4
  float* a_dst   = a_src + (size_t)NNODES * NHEADS;      // N*4
  unsigned* segk = (unsigned*)(a_dst + (size_t)NNODES * NHEADS); // N*4
  float* denom   = (float*)segk + (size_t)NNODES * NHEADS;       // N*4
  float* alpha   = denom + (size_t)NNODES * NHEADS;      // (E+N)*4

  gat_init<<<(NNODES * DCH + 255) / 256, 256, 0, stream>>>(out, denom, segk);
  gat_gemm_wmma<<<NNODES / 16, 256, 0, stream>>>(x, W, xh);
  gat_logits<<<NNODES / 8, 256, 0, stream>>>(xh, att_src, att_dst, a_src, a_dst);
  gat_edge_score<<<(NTOT + 255) / 256, 256, 0, stream>>>(ei, a_src, a_dst, alpha, segk);
  gat_edge_exp<<<(NTOT + 255) / 256, 256, 0, stream>>>(ei, segk, alpha, denom);
  gat_aggregate<<<(NTOT + 7) / 8, 256, 0, stream>>>(ei, xh, alpha, denom, out);
  gat_bias<<<(NNODES * DCH + 255) / 256, 256, 0, stream>>>(out, bias);
}